// Kmeans_Model_31671088841242
// MI455X (gfx1250) — compile-verified
//
#include <hip/hip_runtime.h>

#define N_PTS   262144
#define DIM     128
#define K_CL    102
#define K_PAD   112            // 7 tiles of 16 centroids
#define N_TILES 7
#define ITERS   10
#define THREADS 256
#define WAVES   8
#define PTS_PER_WG 1024
#define WG_COUNT   (N_PTS / PTS_PER_WG)          // 256 workgroups
#define PASSES     (PTS_PER_WG / (WAVES * 16))   // 8 passes per wave
#define C_STRIDE   132         // padded LDS row stride (floats): 132 mod 64 = 4 -> bank-conflict-free b64 reads

// LDS layout (bytes):
//  phase1: [0,59136) c_lds (112*132 f32), [59136,59584) csq (112 f32)
//  phase2: [0,52224) sums (102*128 f32), [52224,52672) counts (112 f32)
//  both:   [59584,60608) labels u8 (1024)
#define SMEM_BYTES 60608
#define OFF_CSQ    59136
#define OFF_CNT    52224
#define OFF_LAB    59584

typedef float v2f __attribute__((ext_vector_type(2)));
typedef float v8f __attribute__((ext_vector_type(8)));

__global__ void k_init_c(const float* __restrict__ x, float* __restrict__ c) {
    int i = blockIdx.x * blockDim.x + threadIdx.x;
    if (i < K_CL * DIM) c[i] = x[i];
}

__global__ void k_zero(float* __restrict__ sums, float* __restrict__ counts) {
    int i = blockIdx.x * blockDim.x + threadIdx.x;
    if (i < K_CL * DIM) sums[i] = 0.0f;
    if (i < K_CL)       counts[i] = 0.0f;
}

__global__ __launch_bounds__(THREADS, 1)
void k_assign(const float* __restrict__ x, const float* __restrict__ c,
              float* __restrict__ sums, float* __restrict__ counts,
              int* __restrict__ labels) {
    __shared__ __align__(16) char smem[SMEM_BYTES];
    float* c_lds   = (float*)smem;
    float* csq_lds = (float*)(smem + OFF_CSQ);
    float* s_sums  = (float*)smem;
    float* s_cnt   = (float*)(smem + OFF_CNT);
    unsigned char* s_lab = (unsigned char*)(smem + OFF_LAB);

    const int tid = threadIdx.x;
    const int wg_base = blockIdx.x * PTS_PER_WG;

    // ---- load centroids into padded LDS; pad rows are zero ----
    for (int i = tid; i < K_CL * DIM; i += THREADS) {
        int r = i / DIM, col = i - r * DIM;
        c_lds[r * C_STRIDE + col] = c[i];
    }
    for (int i = tid; i < (K_PAD - K_CL) * DIM; i += THREADS) {
        int r = K_CL + i / DIM, col = i % DIM;
        c_lds[r * C_STRIDE + col] = 0.0f;
    }
    __syncthreads();

    // ---- per-centroid squared norms (pad rows get +inf-ish) ----
    if (tid < K_PAD) {
        float s = 1e30f;
        if (tid < K_CL) {
            s = 0.0f;
            for (int d = 0; d < DIM; ++d) {
                float v = c_lds[tid * C_STRIDE + d];
                s = fmaf(v, v, s);
            }
        }
        csq_lds[tid] = s;
    }
    __syncthreads();

    const int lane = tid & 31;
    const int wave = tid >> 5;
    const int l = lane & 15;   // column-in-tile lane
    const int h = lane >> 4;   // K-split half

    for (int pass = 0; pass < PASSES; ++pass) {
        const int lp = (pass * WAVES + wave) * 16;        // local point base
        const float* xrow = x + (size_t)(wg_base + lp + l) * DIM + 2 * h;

        v8f acc[N_TILES] = {};

        // ---- 16x112 dot-product tile via fp32 WMMA, K = 128 in steps of 4 ----
        #pragma unroll 4
        for (int kk = 0; kk < DIM / 4; ++kk) {
            v2f a;
            a.x = xrow[kk * 4 + 0];
            a.y = xrow[kk * 4 + 1];
            #pragma unroll
            for (int t = 0; t < N_TILES; ++t) {
                const float* bp = c_lds + (t * 16 + l) * C_STRIDE + kk * 4 + 2 * h;
                v2f b;
                b.x = bp[0];
                b.y = bp[1];
                acc[t] = __builtin_amdgcn_wmma_f32_16x16x4_f32(
                    false, a, false, b, (short)0, acc[t], false, false);
            }
        }

        // ---- argmin over centroids: d2-equivalent = csq - 2*dot ----
        #pragma unroll
        for (int j = 0; j < 8; ++j) {
            float bestv = 3.4e38f;
            int   besti = 0;
            #pragma unroll
            for (int t = 0; t < N_TILES; ++t) {
                float v = fmaf(-2.0f, acc[t][j], csq_lds[l + 16 * t]);
                int   n = l + 16 * t;
                if (v < bestv || (v == bestv && n < besti)) { bestv = v; besti = n; }
            }
            // butterfly within each 16-lane half (masks < 16 never cross halves)
            #pragma unroll
            for (int mk = 1; mk < 16; mk <<= 1) {
                float ov = __shfl_xor(bestv, mk, 32);
                int   oi = __shfl_xor(besti, mk, 32);
                if (ov < bestv || (ov == bestv && oi < besti)) { bestv = ov; besti = oi; }
            }
            if (l == j) {  // one lane per (j,h) writes point lp + j + 8*h
                int pl = lp + j + 8 * h;
                labels[wg_base + pl] = besti;
                s_lab[pl] = (unsigned char)besti;
            }
        }
    }

    __syncthreads();   // phase 1 done; LDS region is reused below

    // ---- phase 2: per-WG segment sums in LDS ----
    for (int i = tid; i < K_CL * DIM; i += THREADS) s_sums[i] = 0.0f;
    for (int i = tid; i < K_PAD; i += THREADS)      s_cnt[i]  = 0.0f;
    __syncthreads();

    for (int p = wave; p < PTS_PER_WG; p += WAVES) {
        int lab = (int)s_lab[p];                      // LDS broadcast
        const float* xp = x + (size_t)(wg_base + p) * DIM;
        float* sp = s_sums + lab * DIM;
        #pragma unroll
        for (int i = 0; i < 4; ++i) {
            float v = xp[lane + 32 * i];              // coalesced b32 loads
            atomicAdd(&sp[lane + 32 * i], v);         // ds_add_f32, consecutive banks
        }
        if (lane == 0) atomicAdd(&s_cnt[lab], 1.0f);
    }
    __syncthreads();

    // ---- flush to global with float atomics ----
    for (int i = tid; i < K_CL * DIM; i += THREADS)
        unsafeAtomicAdd(&sums[i], s_sums[i]);
    if (tid < K_CL)
        unsafeAtomicAdd(&counts[tid], s_cnt[tid]);
}

__global__ void k_update(const float* __restrict__ sums,
                         const float* __restrict__ counts,
                         float* __restrict__ c) {
    int i = blockIdx.x;     // cluster
    int d = threadIdx.x;    // dim
    float cnt = counts[i];
    float v = c[i * DIM + d];
    if (cnt > 0.0f) v = sums[i * DIM + d] / cnt;   // max(cnt,1)==cnt when cnt>0
    c[i * DIM + d] = v;
}

extern "C" void kernel_launch(void* const* d_in, const int* in_sizes, int n_in,
                              void* d_out, int out_size, void* d_ws, size_t ws_size,
                              hipStream_t stream) {
    const float* x = (const float*)d_in[0];
    int* labels = (int*)d_out;

    char* ws = (char*)d_ws;
    float* d_c      = (float*)(ws);            // 52224 B
    float* d_sums   = (float*)(ws + 65536);    // 52224 B
    float* d_counts = (float*)(ws + 131072);   // 408 B

    k_init_c<<<(K_CL * DIM + 255) / 256, 256, 0, stream>>>(x, d_c);
    for (int it = 0; it < ITERS; ++it) {
        k_zero<<<(K_CL * DIM + 255) / 256, 256, 0, stream>>>(d_sums, d_counts);
        k_assign<<<WG_COUNT, THREADS, 0, stream>>>(x, d_c, d_sums, d_counts, labels);
        k_update<<<K_CL, DIM, 0, stream>>>(d_sums, d_counts, d_c);
    }
}